// VqGanClipGenerator_72670846648387
// MI455X (gfx1250) — compile-verified
//
#include <hip/hip_runtime.h>
#include <hip/hip_bf16.h>

typedef float v2f __attribute__((ext_vector_type(2)));
typedef float v8f __attribute__((ext_vector_type(8)));
typedef unsigned int v4u __attribute__((ext_vector_type(4)));
typedef int v4i __attribute__((ext_vector_type(4)));
typedef int v8i __attribute__((ext_vector_type(8)));

// LDS byte offset of a __shared__ object (generic-pointer low 32 bits).
__device__ __forceinline__ unsigned lds_off(const void* p) {
    return (unsigned)(unsigned long long)(uintptr_t)p;
}

// ---------------------------------------------------------------------------
// Tensor Data Mover: 2D tile load Global -> LDS.
//   dim0/dim1     : tensor extents (data_size=4B units)
//   tile0/tile1   : tile extents
//   stride0       : elements between consecutive dim-1 lines
//   padBits       : pad_enable/pad_interval/pad_amount bits for D# group1 word0
// ---------------------------------------------------------------------------
__device__ __forceinline__ void tdm_load_2d(
    unsigned ldsAddr, const void* gptr,
    unsigned dim0, unsigned dim1, unsigned tile0, unsigned tile1,
    unsigned long long stride0, unsigned padBits)
{
    const unsigned long long ga = (unsigned long long)(uintptr_t)gptr;
    v4u g0;
    g0.x = 1u;                                                  // count=1 (valid user D#)
    g0.y = ldsAddr;                                             // lds_addr (bytes)
    g0.z = (unsigned)ga;                                        // global_addr[31:0]
    g0.w = (unsigned)((ga >> 32) & 0x1FFFFFFull) | (2u << 30);  // addr[56:32] | type=2

    v8i g1;
    g1[0] = (int)((2u << 16) | padBits);                 // data_size=4B | pad cfg
    g1[1] = (int)(dim0 << 16);                           // tensor_dim0[15:0]
    g1[2] = (int)((dim0 >> 16) | (dim1 << 16));          // dim0[31:16] | dim1[15:0]
    g1[3] = (int)((dim1 >> 16) | (tile0 << 16));         // dim1[31:16] | tile_dim0
    g1[4] = (int)(tile1 & 0xFFFFu);                      // tile_dim1 | tile_dim2=0
    g1[5] = (int)(unsigned)(stride0 & 0xFFFFFFFFull);    // dim0_stride[31:0]
    g1[6] = (int)(unsigned)((stride0 >> 32) & 0xFFFFull);// dim0_stride[47:32]
    g1[7] = 0;

    const v4i g2 = {0, 0, 0, 0};
    const v4i g3 = {0, 0, 0, 0};
#if __clang_major__ >= 23
    const v8i g4 = {0, 0, 0, 0, 0, 0, 0, 0};
    __builtin_amdgcn_tensor_load_to_lds(g0, g1, g2, g3, g4, 0);
#else
    __builtin_amdgcn_tensor_load_to_lds(g0, g1, g2, g3, 0);
#endif
}

// ---------------------------------------------------------------------------
// Kernel 1: codebook squared norms + init of global argmin keys.
// ---------------------------------------------------------------------------
__global__ __launch_bounds__(256) void enorm_init_kernel(
    const float* __restrict__ cb, float* __restrict__ eNorm,
    unsigned long long* __restrict__ wsKey)
{
    const int gtid = blockIdx.x * 256 + threadIdx.x;
    if (gtid < 4096) wsKey[gtid] = ~0ull;

    const int lane = threadIdx.x & 31;
    const int row  = blockIdx.x * 8 + (threadIdx.x >> 5);
    const float4* src = (const float4*)(cb + (long)row * 256);
    float4 p0 = src[lane];
    float4 p1 = src[lane + 32];
    float s = p0.x*p0.x + p0.y*p0.y + p0.z*p0.z + p0.w*p0.w
            + p1.x*p1.x + p1.y*p1.y + p1.z*p1.z + p1.w*p1.w;
#pragma unroll
    for (int m = 16; m >= 1; m >>= 1) s += __shfl_xor(s, m, 32);
    if (lane == 0) eNorm[row] = s;
}

// ---------------------------------------------------------------------------
// Kernel 2: f32 WMMA GEMM scores + running argmin.
// Block: 128 threads = 4 waves. Block tile: 32 rows x 64 codebook entries.
// Wave (rTile, cHalf): 16 rows x 32 cols = 2 accumulators, A-frag reused.
// A^T tile (ldsA[k][m]) and padded B tile staged by the Tensor Data Mover.
// Grid: (128 M-tiles, 8 N-splits of 2048).
// ---------------------------------------------------------------------------
__global__ __launch_bounds__(128) void vq_argmin_kernel(
    const float* __restrict__ z, const float* __restrict__ cb,
    const float* __restrict__ eNorm, unsigned long long* __restrict__ wsKey)
{
    __shared__ float ldsA[256][32];      // A^T: [k][m], conflict-free frag reads
    __shared__ float ldsB[64 * 65];      // 64 rows x 64 K-floats + 1 TDM pad float
    __shared__ unsigned long long ldsMin[32];

    const int t     = threadIdx.x;
    const int lane  = t & 31;
    const int wave  = t >> 5;
    const int rTile = wave & 1;                  // 16-row tile
    const int cHalf = wave >> 1;                 // 32-col half of the 64-col chunk
    const int lc    = lane & 15;
    const int kp    = (lane >> 4) << 1;          // K-pair select (0 or 2)

    const int mBase = blockIdx.x * 32;           // fixed (b, h), w = 0..31
    const int b     = mBase >> 10;
    const int h     = (mBase >> 5) & 31;

    if (t < 32) ldsMin[t] = ~0ull;

    // TDM: stage A^T = z[b, 0:256, h, 0:32] -> ldsA[c][w]; rows of 32 floats.
    if (wave == 0) {
        tdm_load_2d(lds_off(&ldsA[0][0]),
                    z + (size_t)b * 262144 + (size_t)h * 32,
                    /*dim0*/1024, /*dim1*/256, /*tile0*/32, /*tile1*/256,
                    /*stride0*/1024, /*pad*/0u);
        __builtin_amdgcn_s_wait_tensorcnt((short)0);
    }

    const int aRow  = rTile * 16 + lc;
    const int nBase = blockIdx.y * 2048;
    const unsigned ldsBoff = lds_off(&ldsB[0]);
    // pad_enable | interval code 5 (64 DWORDs) | amount code 0 (1 DWORD)
    const unsigned padB = (1u << 20) | (5u << 22) | (0u << 25);

    float rv0[8], rv1[8]; int ri0[8], ri1[8];
#pragma unroll
    for (int v = 0; v < 8; ++v) {
        rv0[v] = 3.4e38f; rv1[v] = 3.4e38f; ri0[v] = 0; ri1[v] = 0;
    }

    for (int n0 = nBase; n0 < nBase + 2048; n0 += 64) {
        v8f acc0, acc1;
#pragma unroll
        for (int i = 0; i < 8; ++i) { acc0[i] = 0.0f; acc1[i] = 0.0f; }

        for (int kq = 0; kq < 4; ++kq) {
            __syncthreads();                      // prior quarter fully consumed
            if (wave == 0) {
                // 64 codebook rows x 64 K-floats, padded to 65-float LDS rows.
                tdm_load_2d(ldsBoff,
                            cb + (size_t)n0 * 256 + (size_t)kq * 64,
                            /*dim0*/256, /*dim1*/16384, /*tile0*/64, /*tile1*/64,
                            /*stride0*/256, padB);
                __builtin_amdgcn_s_wait_tensorcnt((short)0);
            }
            __syncthreads();                      // B tile visible to all waves

            const int kO = kq * 64;
#pragma unroll
            for (int kk = 0; kk < 64; kk += 4) {
                v2f a, b0, b1;
                a.x = ldsA[kO + kk + kp][aRow];
                a.y = ldsA[kO + kk + kp + 1][aRow];
                const float* pb = &ldsB[(cHalf * 32 + lc) * 65 + kk + kp];
                b0.x = pb[0];       b0.y = pb[1];
                b1.x = pb[16 * 65]; b1.y = pb[16 * 65 + 1];
                acc0 = __builtin_amdgcn_wmma_f32_16x16x4_f32(
                           false, a, false, b0, (short)0, acc0, false, false);
                acc1 = __builtin_amdgcn_wmma_f32_16x16x4_f32(
                           false, a, false, b1, (short)0, acc1, false, false);
            }
        }

        // score = ||e||^2 - 2 x.e ; running min per (row, lane-col)
        const int nA = n0 + cHalf * 32 + lc;
        const int nB = nA + 16;
        const float enA = eNorm[nA];
        const float enB = eNorm[nB];
#pragma unroll
        for (int v = 0; v < 8; ++v) {
            const float s0 = __builtin_fmaf(-2.0f, acc0[v], enA);
            const float s1 = __builtin_fmaf(-2.0f, acc1[v], enB);
            if (s0 < rv0[v]) { rv0[v] = s0; ri0[v] = nA; }
            if (s1 < rv1[v]) { rv1[v] = s1; ri1[v] = nB; }
        }
    }

    // Order-preserving float->u32 map, pack with index, ds_min_u64 reduce.
#pragma unroll
    for (int v = 0; v < 8; ++v) {
        const int mLocal = rTile * 16 + v + ((lane >> 4) << 3);
        unsigned u0 = __float_as_uint(rv0[v]);
        u0 = (u0 & 0x80000000u) ? ~u0 : (u0 | 0x80000000u);
        atomicMin(&ldsMin[mLocal],
                  ((unsigned long long)u0 << 32) | (unsigned)ri0[v]);
        unsigned u1 = __float_as_uint(rv1[v]);
        u1 = (u1 & 0x80000000u) ? ~u1 : (u1 | 0x80000000u);
        atomicMin(&ldsMin[mLocal],
                  ((unsigned long long)u1 << 32) | (unsigned)ri1[v]);
    }
    __syncthreads();
    if (t < 32) atomicMin(&wsKey[mBase + t], ldsMin[t]);   // cross N-split reduce
}

// ---------------------------------------------------------------------------
// Kernel 3: gather z_q[b,c,h,w] = codebook[idx[m], c]; flat out index == tid.
// ---------------------------------------------------------------------------
__global__ __launch_bounds__(256) void gather_kernel(
    const float* __restrict__ cb, const unsigned long long* __restrict__ wsKey,
    float* __restrict__ out)
{
    const int t = blockIdx.x * 256 + threadIdx.x;    // 0..1048575
    const int w = t & 31;
    const int h = (t >> 5) & 31;
    const int c = (t >> 10) & 255;
    const int b = t >> 18;
    const int m = b * 1024 + h * 32 + w;
    const int idx = (int)(wsKey[m] & 0xFFFFFFFFull);
    out[t] = cb[(long)idx * 256 + c];
}

// ---------------------------------------------------------------------------
// Kernel 4: spherical_dist_loss (axis=1 normalization, faithful to reference).
// ---------------------------------------------------------------------------
__global__ __launch_bounds__(256) void sph_loss_kernel(
    const float* __restrict__ enc, const float* __restrict__ prm,
    float* __restrict__ out)
{
    const int e = blockIdx.x * 256 + threadIdx.x;
    if (e >= 512) return;
    float yv[4]; float yn2 = 0.0f;
#pragma unroll
    for (int b = 0; b < 4; ++b) { yv[b] = prm[b * 512 + e]; yn2 += yv[b] * yv[b]; }
    const float yinv = 1.0f / fmaxf(sqrtf(yn2), 1e-12f);

    float accm = 0.0f;
    for (int i = 0; i < 32; ++i) {
        float xv[4]; float xn2 = 0.0f;
#pragma unroll
        for (int b = 0; b < 4; ++b) {
            xv[b] = enc[(i * 4 + b) * 512 + e];
            xn2 += xv[b] * xv[b];
        }
        const float xinv = 1.0f / fmaxf(sqrtf(xn2), 1e-12f);
        float ds = 0.0f;
#pragma unroll
        for (int b = 0; b < 4; ++b) {
            const float dd = xv[b] * xinv - yv[b] * yinv;
            ds += dd * dd;
        }
        const float a = asinf(0.5f * sqrtf(ds));
        accm += 2.0f * a * a;
    }
    out[1048576 + e] = accm * (1.0f / 32.0f);
}

// ---------------------------------------------------------------------------
extern "C" void kernel_launch(void* const* d_in, const int* in_sizes, int n_in,
                              void* d_out, int out_size, void* d_ws, size_t ws_size,
                              hipStream_t stream) {
    const float* z   = (const float*)d_in[0];   // [4,256,32,32]
    const float* cb  = (const float*)d_in[1];   // [16384,256]
    const float* enc = (const float*)d_in[2];   // [32,4,512]
    const float* prm = (const float*)d_in[3];   // [4,512]
    float* out = (float*)d_out;                 // [1048576 z_q | 512 dists]

    char* ws = (char*)d_ws;
    unsigned long long* wsKey = (unsigned long long*)ws;              // 4096 * 8B
    float* eNorm = (float*)(ws + 4096 * sizeof(unsigned long long));  // 16384 * 4B

    enorm_init_kernel<<<2048, 256, 0, stream>>>(cb, eNorm, wsKey);
    vq_argmin_kernel<<<dim3(128, 8), 128, 0, stream>>>(z, cb, eNorm, wsKey);
    gather_kernel<<<4096, 256, 0, stream>>>(cb, wsKey, out);
    sph_loss_kernel<<<2, 256, 0, stream>>>(enc, prm, out);
}